// DeformLoss_35330400977029
// MI455X (gfx1250) — compile-verified
//
#include <hip/hip_runtime.h>

typedef __attribute__((ext_vector_type(2))) float v2f;
typedef __attribute__((ext_vector_type(8))) float v8f;

#define NPTS 8192
#define NB 4
#define ROWS_PER_BLOCK 128                  // 8 waves x 16 rows
#define ROW_BLOCKS (NPTS / ROWS_PER_BLOCK)  // 64

// ---------------------------------------------------------------------------
// Zero the accumulators in workspace (d_ws is poisoned to 0xAA by the harness)
// ---------------------------------------------------------------------------
__global__ void init_acc_kernel(float* acc) {
    if (threadIdx.x < 4) acc[threadIdx.x] = 0.0f;
}

// ---------------------------------------------------------------------------
// One Chamfer direction: for each point p_i of P (= template + dispP),
//   min_j || p_i - q_j ||^2  over Q (= template + dispQ),
// summed over i into *acc (one f32 atomic per wave).
//
// Identity:  min_j ||p-q||^2 = ||p||^2 + 2 * min_j( 0.5*||q||^2 - p.q )
//
// The whole per-pair term (0.5*||q||^2 - p.q) comes out of ONE
// v_wmma_f32_16x16x4_f32 per 16x16 tile by exploiting the K=4 padding slot:
//   A row m  = [ -px, -py, -pz, 1.0 ]        (16x4, striped per ISA layout)
//   B col n  = [  qx,  qy,  qz, 0.5*|q|^2 ]  (4x16)
//   C        = 0 (inline/hoisted constant)
// so D[m][n] = 0.5*|q_n|^2 - p_m.q_n with zero per-tile VALU setup.
// ---------------------------------------------------------------------------
__global__ __launch_bounds__(256)
void chamfer_rowmin_kernel(const float* __restrict__ dispP,
                           const float* __restrict__ dispQ,
                           const float* __restrict__ tmpl,
                           float* __restrict__ acc)
{
    // Two NPTS-long v2f arrays: [0,NPTS) = (x,y), [NPTS,2*NPTS) = (z, .5|q|^2)
    extern __shared__ v2f qsmem[];           // 2 * 8192 * 8B = 128 KB
    v2f* qs_xy = qsmem;
    v2f* qs_zw = qsmem + NPTS;

    const int  tid  = threadIdx.x;
    const int  lane = tid & 31;
    const int  wave = tid >> 5;
    const int  b    = blockIdx.x >> 6;        // ROW_BLOCKS == 64
    const int  rb   = blockIdx.x & 63;
    const long base = (long)b * NPTS;

    // ---- stage all Q points into LDS cooperatively ----
    for (int i = tid; i < NPTS; i += 256) {
        const long o = (base + i) * 3;
        float qx = tmpl[o + 0] + dispQ[o + 0];
        float qy = tmpl[o + 1] + dispQ[o + 1];
        float qz = tmpl[o + 2] + dispQ[o + 2];
        v2f xy; xy.x = qx; xy.y = qy;
        v2f zw; zw.x = qz; zw.y = 0.5f * (qx * qx + qy * qy + qz * qz);
        qs_xy[i] = xy;
        qs_zw[i] = zw;
    }
    __syncthreads();

    // ---- each wave owns 16 rows of P; load (negated) P point ----
    const int  m  = rb * ROWS_PER_BLOCK + wave * 16 + (lane & 15);
    const long po = (base + m) * 3;
    float px = -(tmpl[po + 0] + dispP[po + 0]);
    float py = -(tmpl[po + 1] + dispP[po + 1]);
    float pz = -(tmpl[po + 2] + dispP[po + 2]);
    float pnorm = px * px + py * py + pz * pz;

    // A operand (16x4 f32): lanes 0-15 carry (k0,k1), lanes 16-31 (k2,k3)
    v2f a;
    a.x = (lane < 16) ? px : pz;
    a.y = (lane < 16) ? py : 1.0f;     // k3 = 1.0 multiplies B's 0.5*|q|^2 row

    // B operand source: lo lanes read (x,y), hi lanes read (z, 0.5|q|^2);
    // select the pointer ONCE -- per tile it's a single ds_load_b64.
    const v2f* src = (lane < 16) ? qs_xy : qs_zw;
    const int  col = lane & 15;

    v8f zero;
#pragma unroll
    for (int r = 0; r < 8; ++r) zero[r] = 0.0f;   // loop-invariant C

    v8f rmin;
#pragma unroll
    for (int r = 0; r < 8; ++r) rmin[r] = 3.0e38f;

#pragma unroll 8
    for (int t = 0; t < NPTS / 16; ++t) {
        v2f bb = src[t * 16 + col];               // ds_load_b64 == B operand
        v8f c = __builtin_amdgcn_wmma_f32_16x16x4_f32(
                    /*neg_a=*/false, a, /*neg_b=*/false, bb,
                    /*c_mod=*/(short)0, zero,
                    /*reuse_a=*/false, /*reuse_b=*/false);
#pragma unroll
        for (int r = 0; r < 8; ++r) rmin[r] = fminf(rmin[r], c[r]);
    }

    // ---- cross-lane min within each 16-lane half (lanes of a half hold the same 8 M rows) ----
#pragma unroll
    for (int off = 1; off <= 8; off <<= 1) {
#pragma unroll
        for (int r = 0; r < 8; ++r)
            rmin[r] = fminf(rmin[r], __shfl_xor(rmin[r], off, 32));
    }

    // sum of the 8 per-half row minima, then merge halves -> 16 rows total
    float s = rmin[0] + rmin[1] + rmin[2] + rmin[3] +
              rmin[4] + rmin[5] + rmin[6] + rmin[7];
    s += __shfl_xor(s, 16, 32);

    // sum of |p|^2 over the wave's 16 rows (lanes 0-15 hold rows 0..15;
    // hi half holds duplicates, reduction stays inside each half)
    float ps = pnorm;
#pragma unroll
    for (int off = 1; off <= 8; off <<= 1) ps += __shfl_xor(ps, off, 32);

    if (lane == 0) {
        float partial = ps + 2.0f * s;   // sum_i min_j ||p_i - q_j||^2
        __hip_atomic_fetch_add(acc, partial, __ATOMIC_RELAXED,
                               __HIP_MEMORY_SCOPE_AGENT);
    }
}

// ---------------------------------------------------------------------------
// sum |a-b| over n elements -> atomic accumulate into *acc
// ---------------------------------------------------------------------------
__global__ __launch_bounds__(256)
void l1_reduce_kernel(const float* __restrict__ a, const float* __restrict__ b,
                      float* __restrict__ acc, int n)
{
    float s = 0.0f;
    for (int i = blockIdx.x * blockDim.x + threadIdx.x; i < n;
         i += gridDim.x * blockDim.x)
        s += fabsf(a[i] - b[i]);
#pragma unroll
    for (int off = 1; off < 32; off <<= 1) s += __shfl_xor(s, off, 32);
    if ((threadIdx.x & 31) == 0)
        __hip_atomic_fetch_add(acc, s, __ATOMIC_RELAXED,
                               __HIP_MEMORY_SCOPE_AGENT);
}

// ---------------------------------------------------------------------------
// total = 1.0*mean|dd| + 0.1*mean((pm-tm)^2) + 0.5*(accP2T+accT2P)/(B*N)
// ---------------------------------------------------------------------------
__global__ void finalize_kernel(const float* __restrict__ pred_mat,
                                const float* __restrict__ tgt_mat,
                                const float* __restrict__ acc,
                                float* __restrict__ out,
                                int n_disp, int n_mat)
{
    float mat = 0.0f;
    for (int i = 0; i < n_mat; ++i) {
        float d = pred_mat[i] - tgt_mat[i];
        mat += d * d;
    }
    mat /= (float)n_mat;

    float disp = acc[2] / (float)n_disp;
    float cd   = (acc[0] + acc[1]) / (float)(NB * NPTS);

    out[0] = 1.0f * disp + 0.1f * mat + 0.5f * cd;
}

extern "C" void kernel_launch(void* const* d_in, const int* in_sizes, int n_in,
                              void* d_out, int out_size, void* d_ws, size_t ws_size,
                              hipStream_t stream)
{
    const float* pred_disp = (const float*)d_in[0];
    const float* pred_mat  = (const float*)d_in[1];
    const float* tgt_disp  = (const float*)d_in[2];
    const float* tgt_mat   = (const float*)d_in[3];
    const float* tmpl      = (const float*)d_in[4];

    float* acc = (float*)d_ws;   // [0]=p2t sum, [1]=t2p sum, [2]=L1 sum

    init_acc_kernel<<<1, 32, 0, stream>>>(acc);

    const dim3   grid(NB * ROW_BLOCKS);                  // 256 workgroups
    const size_t lds_bytes = (size_t)2 * NPTS * sizeof(v2f);  // 128 KB

    // pred -> target direction, then target -> pred
    chamfer_rowmin_kernel<<<grid, 256, lds_bytes, stream>>>(
        pred_disp, tgt_disp, tmpl, acc + 0);
    chamfer_rowmin_kernel<<<grid, 256, lds_bytes, stream>>>(
        tgt_disp, pred_disp, tmpl, acc + 1);

    l1_reduce_kernel<<<96, 256, 0, stream>>>(pred_disp, tgt_disp, acc + 2,
                                             in_sizes[0]);

    finalize_kernel<<<1, 1, 0, stream>>>(pred_mat, tgt_mat, acc, (float*)d_out,
                                         in_sizes[0], in_sizes[1]);
}